// PyQHamiltonianEvolution_53300544143778
// MI455X (gfx1250) — compile-verified
//
#include <hip/hip_runtime.h>

#define DIM 8
#define BATCH 8

typedef float v2f __attribute__((ext_vector_type(2)));
typedef float v8f __attribute__((ext_vector_type(8)));

// ---------------------------------------------------------------------------
// Kernel 1: eigh(H) via complex Jacobi, build U_b = V diag(e^{-i t_b w}) V^H,
// pack real 16x16 matrices M_b = [[Ur,-Ui],[Ui,Ur]] row-major into ws.
// Tiny one-off work: single thread.
// ---------------------------------------------------------------------------
__global__ void build_unitaries(const float* __restrict__ hre,
                                const float* __restrict__ him,
                                const float* __restrict__ t,
                                float* __restrict__ M)
{
    if (threadIdx.x != 0) return;

    float Ar[DIM][DIM], Ai[DIM][DIM], Vr[DIM][DIM], Vi[DIM][DIM];
    for (int i = 0; i < DIM; ++i)
        for (int j = 0; j < DIM; ++j) {
            Ar[i][j] = hre[i * DIM + j];
            Ai[i][j] = him[i * DIM + j];
            Vr[i][j] = (i == j) ? 1.f : 0.f;
            Vi[i][j] = 0.f;
        }

    // Complex Jacobi sweeps
    for (int sweep = 0; sweep < 30; ++sweep) {
        for (int p = 0; p < DIM - 1; ++p) {
            for (int q = p + 1; q < DIM; ++q) {
                float zr = Ar[p][q], zi = Ai[p][q];
                float az = sqrtf(zr * zr + zi * zi);
                if (az < 1e-12f) continue;
                float phr = zr / az, phi = zi / az;   // e^{i phi}
                float app = Ar[p][p], aqq = Ar[q][q];
                float tau = (aqq - app) / (2.f * az);
                float tt  = (tau >= 0.f ? 1.f : -1.f) /
                            (fabsf(tau) + sqrtf(1.f + tau * tau));
                float c = 1.f / sqrtf(1.f + tt * tt);
                float s = tt * c;
                // rows: new_p = c*row_p - s*e^{i phi}*row_q ;
                //       new_q = s*row_p + c*e^{i phi}*row_q
                for (int r = 0; r < DIM; ++r) {
                    float pr = Ar[p][r], pi = Ai[p][r];
                    float qr = Ar[q][r], qi = Ai[q][r];
                    float eqr = phr * qr - phi * qi;
                    float eqi = phr * qi + phi * qr;
                    Ar[p][r] = c * pr - s * eqr;  Ai[p][r] = c * pi - s * eqi;
                    Ar[q][r] = s * pr + c * eqr;  Ai[q][r] = s * pi + c * eqi;
                }
                // cols: new_p = c*col_p - s*e^{-i phi}*col_q ;
                //       new_q = s*col_p + c*e^{-i phi}*col_q   (V same)
                for (int r = 0; r < DIM; ++r) {
                    float pr = Ar[r][p], pi = Ai[r][p];
                    float qr = Ar[r][q], qi = Ai[r][q];
                    float eqr = phr * qr + phi * qi;
                    float eqi = phr * qi - phi * qr;
                    Ar[r][p] = c * pr - s * eqr;  Ai[r][p] = c * pi - s * eqi;
                    Ar[r][q] = s * pr + c * eqr;  Ai[r][q] = s * pi + c * eqi;

                    pr = Vr[r][p]; pi = Vi[r][p];
                    qr = Vr[r][q]; qi = Vi[r][q];
                    eqr = phr * qr + phi * qi;
                    eqi = phr * qi - phi * qr;
                    Vr[r][p] = c * pr - s * eqr;  Vi[r][p] = c * pi - s * eqi;
                    Vr[r][q] = s * pr + c * eqr;  Vi[r][q] = s * pi + c * eqi;
                }
            }
        }
    }

    float w[DIM];
    for (int k = 0; k < DIM; ++k) w[k] = Ar[k][k];

    for (int b = 0; b < BATCH; ++b) {
        float tb = t[b];
        float* Mb = M + b * 256;
        for (int i = 0; i < DIM; ++i) {
            for (int j = 0; j < DIM; ++j) {
                float ur = 0.f, ui = 0.f;
                for (int k = 0; k < DIM; ++k) {
                    float cp = cosf(tb * w[k]);
                    float sp = -sinf(tb * w[k]);     // e^{-i tb w}
                    float a = Vr[i][k], bb = Vi[i][k];
                    float cc = Vr[j][k], dd = Vi[j][k];
                    // V[i][k] * conj(V[j][k])
                    float pr = a * cc + bb * dd;
                    float pi = bb * cc - a * dd;
                    ur += pr * cp - pi * sp;
                    ui += pr * sp + pi * cp;
                }
                Mb[i * 16 + j]            = ur;   // Ur
                Mb[i * 16 + (j + 8)]      = -ui;  // -Ui
                Mb[(i + 8) * 16 + j]      = ui;   // Ui
                Mb[(i + 8) * 16 + (j + 8)] = ur;  // Ur
            }
        }
    }
}

// ---------------------------------------------------------------------------
// Kernel 2: apply U_b to state via fp32 WMMA (16x16x4), LDS-staged for
// coalescing. One block = one "group": 8 support-chunks x 128 contiguous
// floats x {re,im}. 8 waves/block, wave w handles batch w (16 columns).
// ---------------------------------------------------------------------------
__global__ __launch_bounds__(256) void apply_unitary(
    const float* __restrict__ sre, const float* __restrict__ sim,
    const float* __restrict__ M, float* __restrict__ out)
{
    __shared__ float lin [2][8][128];   // [plane re/im][support idx][rlow*8+batch]
    __shared__ float lout[2][8][128];

    const int tid = threadIdx.x;
    const int g   = blockIdx.x;         // 0..8191 groups

    // group base: g bits 0-5 -> q14..q9 (strides 256..8192 floats),
    //             bits 6-9  -> q7..q4  (32768..262144),
    //             bits 10-12-> q2..q0  (1048576..4194304)
    const long base = ((long)(g & 63) << 8) +
                      ((long)((g >> 6) & 15) << 15) +
                      ((long)((g >> 10) & 7) << 20);

    // cooperative coalesced load: 512 float4 rows, 2 per thread
    for (int i = 0; i < 2; ++i) {
        int idx   = tid + i * 256;          // 0..511
        int plane = idx >> 8;
        int s     = (idx >> 5) & 7;         // support idx = b3*4 + b8*2 + b15
        int o4    = idx & 31;
        long soff = ((long)(s >> 2) << 19) + ((long)((s >> 1) & 1) << 14) +
                    ((long)(s & 1) << 7);
        const float4 v =
            *((const float4*)((plane ? sim : sre) + base + soff) + o4);
        *(float4*)&lin[plane][s][o4 * 4] = v;
    }
    __syncthreads();

    const int lane = tid & 31;
    const int wave = tid >> 5;          // batch index 0..7
    const int col  = lane & 15;         // N / M-row for A
    const int hi   = lane >> 4;         // lane half

    // A fragments (M_b columns), 4 K-chunks of v2f
    const float* Mb = M + wave * 256;
    v2f a[4];
    for (int kk = 0; kk < 4; ++kk) {
        int k0 = kk * 4 + 2 * hi;
        a[kk][0] = Mb[col * 16 + k0];
        a[kk][1] = Mb[col * 16 + k0 + 1];
    }

    v8f c = {};
    for (int kk = 0; kk < 4; ++kk) {
        int k0 = kk * 4 + 2 * hi;
        v2f bf;
        int k = k0;
        bf[0] = lin[k >> 3][k & 7][col * 8 + wave];
        k = k0 + 1;
        bf[1] = lin[k >> 3][k & 7][col * 8 + wave];
        // D(16x16) += A(16x4) x B(4x16), fp32
        c = __builtin_amdgcn_wmma_f32_16x16x4_f32(
                false, a[kk], false, bf, (short)0, c, false, false);
    }

    // D writeback: lanes 0-15 hold real rows (M=r), lanes 16-31 imag (M=r+8)
    for (int r = 0; r < 8; ++r)
        lout[hi][r][col * 8 + wave] = c[r];
    __syncthreads();

    // cooperative store, interleaved complex64, 1KB contiguous runs
    for (int i = 0; i < 4; ++i) {
        int idx = tid + i * 256;            // 0..1023 complex elems
        int s   = idx >> 7;
        int off = idx & 127;
        long soff = ((long)(s >> 2) << 19) + ((long)((s >> 1) & 1) << 14) +
                    ((long)(s & 1) << 7);
        float2 v = make_float2(lout[0][s][off], lout[1][s][off]);
        *(float2*)(out + 2 * (base + soff + off)) = v;
    }
}

// ---------------------------------------------------------------------------
extern "C" void kernel_launch(void* const* d_in, const int* in_sizes, int n_in,
                              void* d_out, int out_size, void* d_ws, size_t ws_size,
                              hipStream_t stream)
{
    const float* sre = (const float*)d_in[0];   // state_real (2^20 * 8)
    const float* sim = (const float*)d_in[1];   // state_imag
    const float* hre = (const float*)d_in[2];   // h_real (8*8*1)
    const float* him = (const float*)d_in[3];   // h_imag
    const float* t   = (const float*)d_in[4];   // (8,)
    float* outp = (float*)d_out;                // complex64 interleaved
    float* Mws  = (float*)d_ws;                 // 8 * 16 * 16 floats

    build_unitaries<<<1, 32, 0, stream>>>(hre, him, t, Mws);
    apply_unitary<<<8192, 256, 0, stream>>>(sre, sim, Mws, outp);
}